// TriangleRasterizer_65120294142041
// MI455X (gfx1250) — compile-verified
//
#include <hip/hip_runtime.h>
#include <math.h>

// ---------------- problem constants (from reference) ----------------
#define IMG_H 128
#define IMG_W 128
#define NTRI  1024
#define NVERT (NTRI * 3)
#define NEAR_P 0.2f
#define EPS_A  1e-8f
// GAMMA == 1.0 in the reference, so cov**GAMMA == cov (no powf needed).

typedef __attribute__((ext_vector_type(2))) float v2f;
typedef __attribute__((ext_vector_type(8))) float v8f;
typedef __attribute__((ext_vector_type(4))) int   v4i;

// ---------------- CDNA5 feature detection ----------------
// These evaluate to 0 on the host pass (host clang doesn't know amdgcn
// builtins), which conveniently shields host compilation too.
#if defined(__has_builtin)
#  if __has_builtin(__builtin_amdgcn_wmma_f32_16x16x4_f32)
#    define HAVE_WMMA4 1
#  endif
#  if __has_builtin(__builtin_amdgcn_global_load_async_to_lds_b128)
#    define HAVE_ASYNC 1
#  endif
#endif
#ifndef HAVE_WMMA4
#  define HAVE_WMMA4 0
#endif
#ifndef HAVE_ASYNC
#  define HAVE_ASYNC 0
#endif

#if defined(__has_builtin) && __has_builtin(__builtin_amdgcn_s_wait_asynccnt)
#  define WAIT_ASYNC(n) __builtin_amdgcn_s_wait_asynccnt(n)
#else
#  define WAIT_ASYNC(n) asm volatile("s_wait_asynccnt " #n ::: "memory")
#endif

#if HAVE_ASYNC
// builtin signature (from clang diagnostic): (global int4*, lds int4*, off, cpol)
typedef __attribute__((address_space(1))) v4i* gas_v4i_p;
typedef __attribute__((address_space(3))) v4i* las_v4i_p;
#define ASYNC_B128(gp, lp) \
  __builtin_amdgcn_global_load_async_to_lds_b128((gas_v4i_p)(gp), (las_v4i_p)(lp), 0, 0)
#endif

// =====================================================================
// Kernel D (defined first so its asm is the visible snippet):
// fused front-to-back compositing. One thread per pixel, 16x16 tile per
// 256-thread block. Sorted records streamed into LDS in 4 KB chunks via
// async global->LDS (double buffered, ASYNCcnt).
// =====================================================================
#define CHUNK  64                     // records per chunk (64 * 64B = 4 KB)
#define NCHUNK (NTRI / CHUNK)

__global__ void raster_kernel(const float* __restrict__ recS,
                              const float* __restrict__ background,
                              float* __restrict__ out) {
  __shared__ float4 lbuf[2 * CHUNK * 4];   // 8 KB double buffer
  const int t  = threadIdx.x;              // 0..255
  const int px = blockIdx.x * 16 + (t & 15);
  const int py = blockIdx.y * 16 + (t >> 4);
  const float X = (float)px + 0.5f;
  const float Y = (float)py + 0.5f;

  const float4* gsrc = (const float4*)recS;  // NTRI*4 float4

  float T = 1.0f, accR = 0.0f, accG = 0.0f, accB = 0.0f;

  // prologue: issue chunk 0 into buffer 0
#if HAVE_ASYNC
  ASYNC_B128(gsrc + t, &lbuf[t]);
#else
  lbuf[t] = gsrc[t];
#endif

  for (int c = 0; c < NCHUNK; ++c) {
    const int cur = c & 1;
    if (c + 1 < NCHUNK) {
      // prefetch next chunk into the other buffer (safe: it was last read
      // in iteration c-1, which ended with a barrier)
#if HAVE_ASYNC
      ASYNC_B128(gsrc + (c + 1) * CHUNK * 4 + t,
                 &lbuf[(1 - cur) * CHUNK * 4 + t]);
#else
      lbuf[(1 - cur) * CHUNK * 4 + t] = gsrc[(c + 1) * CHUNK * 4 + t];
#endif
    }
#if HAVE_ASYNC
    if (c + 1 < NCHUNK) { WAIT_ASYNC(1); } else { WAIT_ASYNC(0); }
#endif
    __syncthreads();   // all waves' current-chunk LDS data visible

    const float4* R  = &lbuf[cur * CHUNK * 4];
    const float*  Rf = (const float*)R;
    #pragma unroll 4
    for (int k = 0; k < CHUNK; ++k) {
      const float4 q0 = R[k * 4 + 0];
      const float4 q1 = R[k * 4 + 1];
      const float4 q2 = R[k * 4 + 2];
      const float  f2 = Rf[k * 16 + 12];

      const float w0 = fmaf(X, q0.x, fmaf(Y, q0.y, q0.z));
      const float w1 = fmaf(X, q0.w, fmaf(Y, q1.x, q1.y));
      const float w2 = fmaf(X, q1.z, fmaf(Y, q1.w, q2.x));
      const float mn  = fminf(w0, fminf(w1, w2));
      const float cov = fminf(fmaxf(3.0f * mn, 0.0f), 1.0f);  // GAMMA == 1
      const float alpha = fminf(q2.y * cov, 0.999f);
      const float wgt = T * alpha;
      accR = fmaf(wgt, q2.z, accR);
      accG = fmaf(wgt, q2.w, accG);
      accB = fmaf(wgt, f2,   accB);
      T *= (1.0f - alpha);
    }
    __syncthreads();   // done reading buf[cur] before it is refilled
  }

  accR = fmaf(T, background[0], accR);
  accG = fmaf(T, background[1], accG);
  accB = fmaf(T, background[2], accB);

  const int pidx = py * IMG_W + px;
  out[0 * IMG_H * IMG_W + pidx] = accR;
  out[1 * IMG_H * IMG_W + pidx] = accG;
  out[2 * IMG_H * IMG_W + pidx] = accB;
}

// =====================================================================
// Kernel A: projection via V_WMMA_F32_16X16X4_F32.
//   D(16x16) = A(16x4) * B(4x16),  A = [x y z 1] rows for 16 vertices,
//   B cols 0..3 = projmatrix, cols 4..7 = viewmatrix.
//   Output per vertex: p_hom (cols 0..3) and zview (col 6).
// Launch: <<<NVERT/16, 32>>> (one wave per 16 vertices, both paths).
// proj layout: NVERT rows of 6 floats: {phx, phy, phz, phw, zview, pad}
// =====================================================================
__global__ void proj_kernel(const float* __restrict__ vertex,
                            const float* __restrict__ vm,
                            const float* __restrict__ pm,
                            float* __restrict__ proj) {
#if HAVE_WMMA4
  const int wave = blockIdx.x;          // 0..191, 16 vertices each
  const int lane = threadIdx.x;         // 0..31
  const int hi   = lane >> 4;           // lane group
  const int m    = lane & 15;

  // A fragment: 16x4 f32. Lanes 0-15 hold row M=lane, K={0,1}=(x,y);
  // lanes 16-31 hold row M=lane-16, K={2,3}=(z,1).
  const float* vp = vertex + (wave * 16 + m) * 3;
  v2f a;
  a.x = hi ? vp[2] : vp[0];
  a.y = hi ? 1.0f  : vp[1];

  // B fragment: 4x16 f32. V0: K=0 (lanes 0-15) / K=1 (lanes 16-31);
  // V1: K=2 / K=3. Column index N = m. B[k][n<4]=pm[k][n], B[k][4..7]=vm[k][n-4].
  const int n  = m;
  const int k0 = hi ? 1 : 0;
  const int k1 = hi ? 3 : 2;
  v2f b;
  b.x = (n < 4) ? pm[k0 * 4 + n] : ((n < 8) ? vm[k0 * 4 + (n - 4)] : 0.0f);
  b.y = (n < 4) ? pm[k1 * 4 + n] : ((n < 8) ? vm[k1 * 4 + (n - 4)] : 0.0f);

  v8f c = {0.f, 0.f, 0.f, 0.f, 0.f, 0.f, 0.f, 0.f};
  v8f d = __builtin_amdgcn_wmma_f32_16x16x4_f32(
      /*neg_a=*/false, a, /*neg_b=*/false, b,
      /*c_mod=*/(short)0, c, /*reuse_a=*/false, /*reuse_b=*/false);

  // D layout: lane holds column N=n; VGPR i is row M=i (lanes 0-15) or
  // M=i+8 (lanes 16-31).
  if (n < 4) {
    #pragma unroll
    for (int i = 0; i < 8; ++i) {
      const int row = hi ? (i + 8) : i;
      proj[(wave * 16 + row) * 6 + n] = d[i];
    }
  } else if (n == 6) {  // viewmatrix column 2 -> zview
    #pragma unroll
    for (int i = 0; i < 8; ++i) {
      const int row = hi ? (i + 8) : i;
      proj[(wave * 16 + row) * 6 + 4] = d[i];
    }
  }
#else
  // Scalar fallback, identical launch geometry (threads 0..15 do one vertex).
  const int m = threadIdx.x;
  if (m < 16) {
    const int v = blockIdx.x * 16 + m;
    const float* vp = vertex + v * 3;
    const float x = vp[0], y = vp[1], z = vp[2];
    #pragma unroll
    for (int j = 0; j < 4; ++j)
      proj[v * 6 + j] = x * pm[0 * 4 + j] + y * pm[1 * 4 + j] +
                        z * pm[2 * 4 + j] + pm[3 * 4 + j];
    proj[v * 6 + 4] = x * vm[2] + y * vm[6] + z * vm[10] + vm[14];
  }
#endif
}

// =====================================================================
// Kernel B: per-triangle setup. Edge planes pre-multiplied by inv_area,
// validity, radii (output, original order), depth, feature/opacity baked
// into a 64-byte record.
// rec layout (16 f32): {p0x,p0y,p0c, p1x,p1y,p1c, p2x,p2y,p2c, op, f0,f1,f2, 0,0,0}
// =====================================================================
__global__ void setup_kernel(const float* __restrict__ proj,
                             const float* __restrict__ opacity,
                             const float* __restrict__ feature,
                             float* __restrict__ rec,
                             float* __restrict__ depth,
                             int*   __restrict__ radii) {
  const int n = blockIdx.x * blockDim.x + threadIdx.x;
  if (n >= NTRI) return;

  float Pxy[3][2], zv[3];
  #pragma unroll
  for (int k = 0; k < 3; ++k) {
    const float* pr = proj + (n * 3 + k) * 6;
    const float inv = 1.0f / (pr[3] + 1e-7f);
    const float nx = pr[0] * inv, ny = pr[1] * inv;
    Pxy[k][0] = ((nx + 1.0f) * (float)IMG_W - 1.0f) * 0.5f;
    Pxy[k][1] = ((ny + 1.0f) * (float)IMG_H - 1.0f) * 0.5f;
    zv[k] = pr[4];
  }
  const float ax = Pxy[0][0], ay = Pxy[0][1];
  const float bx = Pxy[1][0], by = Pxy[1][1];
  const float cx = Pxy[2][0], cy = Pxy[2][1];

  const float area = (bx - ax) * (cy - ay) - (by - ay) * (cx - ax);
  const bool  areaOK = fabsf(area) > EPS_A;
  const float inv_area = areaOK ? (1.0f / area) : 0.0f;
  const float zmin = fminf(zv[0], fminf(zv[1], zv[2]));
  const bool  valid = areaOK && (zmin > NEAR_P);

  // edge(p,q)(X,Y) = X*(qy-py) - Y*(qx-px) + (py*(qx-px) - px*(qy-py))
  const float p0x = (cy - by) * inv_area;
  const float p0y = -(cx - bx) * inv_area;
  const float p0c = (by * (cx - bx) - bx * (cy - by)) * inv_area;
  const float p1x = (ay - cy) * inv_area;
  const float p1y = -(ax - cx) * inv_area;
  const float p1c = (cy * (ax - cx) - cx * (ay - cy)) * inv_area;
  const float p2x = (by - ay) * inv_area;
  const float p2y = -(bx - ax) * inv_area;
  const float p2c = (ay * (bx - ax) - ax * (by - ay)) * inv_area;

  const float op = valid ? opacity[n] : 0.0f;

  depth[n] = (zv[0] + zv[1] + zv[2]) * (1.0f / 3.0f);

  // radii (original order)
  const float cenx = (ax + bx + cx) * (1.0f / 3.0f);
  const float ceny = (ay + by + cy) * (1.0f / 3.0f);
  float r = 0.0f;
  #pragma unroll
  for (int k = 0; k < 3; ++k) {
    const float dx = Pxy[k][0] - cenx, dy = Pxy[k][1] - ceny;
    r = fmaxf(r, sqrtf(dx * dx + dy * dy));
  }
  radii[n] = valid ? (int)ceilf(r) : 0;

  float4* rp = (float4*)(rec + n * 16);
  rp[0] = make_float4(p0x, p0y, p0c, p1x);
  rp[1] = make_float4(p1y, p1c, p2x, p2y);
  rp[2] = make_float4(p2c, op, feature[n * 3 + 0], feature[n * 3 + 1]);
  rp[3] = make_float4(feature[n * 3 + 2], 0.0f, 0.0f, 0.0f);
}

// =====================================================================
// Kernel C: stable ascending rank by depth (O(N^2) over LDS, N=1024)
// then scatter records into depth order.  <<<1, 1024>>>
// =====================================================================
__global__ void sort_kernel(const float* __restrict__ depth,
                            const float* __restrict__ rec,
                            float* __restrict__ recS) {
  __shared__ float d[NTRI];
  const int i = threadIdx.x;
  d[i] = depth[i];
  __syncthreads();
  const float di = d[i];
  int rank = 0;
  for (int j = 0; j < NTRI; ++j) {
    const float dj = d[j];
    rank += (dj < di) || (dj == di && j < i);  // stable (matches argsort)
  }
  const float4* src = (const float4*)(rec + i * 16);
  float4* dst = (float4*)(recS + rank * 16);
  dst[0] = src[0]; dst[1] = src[1]; dst[2] = src[2]; dst[3] = src[3];
}

// =====================================================================
// Host launcher
// Inputs: 0 vertex(N,3,3) 1 center2D(unused) 2 opacity(N,1) 3 feature(N,3)
//         4 viewmatrix(4,4) 5 projmatrix(4,4) 6 campos(unused) 7 background(3)
// Output: out_feature(3,H,W) f32 then radii(N) i32 (bit-placed after).
// =====================================================================
extern "C" void kernel_launch(void* const* d_in, const int* in_sizes, int n_in,
                              void* d_out, int out_size, void* d_ws, size_t ws_size,
                              hipStream_t stream) {
  const float* vertex  = (const float*)d_in[0];
  const float* opacity = (const float*)d_in[2];
  const float* feature = (const float*)d_in[3];
  const float* vm      = (const float*)d_in[4];
  const float* pm      = (const float*)d_in[5];
  const float* bg      = (const float*)d_in[7];

  float* outF  = (float*)d_out;
  int*   radii = (int*)d_out + 3 * IMG_H * IMG_W;

  // workspace layout (all 16B aligned): ~204 KiB total
  float* ws    = (float*)d_ws;
  float* proj  = ws;                    // NVERT*6  = 18432 floats
  float* rec   = proj + NVERT * 6;      // NTRI*16  = 16384 floats
  float* recS  = rec  + NTRI * 16;      // NTRI*16  = 16384 floats
  float* depth = recS + NTRI * 16;      // NTRI     =  1024 floats

  proj_kernel  <<<NVERT / 16, 32, 0, stream>>>(vertex, vm, pm, proj);
  setup_kernel <<<NTRI / 256, 256, 0, stream>>>(proj, opacity, feature, rec, depth, radii);
  sort_kernel  <<<1, NTRI, 0, stream>>>(depth, rec, recS);
  raster_kernel<<<dim3(IMG_W / 16, IMG_H / 16), 256, 0, stream>>>(recS, bg, outF);
}